// bspline_53163105189888
// MI455X (gfx1250) — compile-verified
//
#include <hip/hip_runtime.h>
#include <stdint.h>

// ---------------------------------------------------------------------------
// B-spline (cubic, K=3) De Boor evaluation over 16.7M points, MI455X/gfx1250.
//   - knots (1024 f32) bitonic-sorted on device into d_ws
//   - sorted knots + 1020 coeffs staged into LDS via global_load_async_to_lds
//   - branchless binary search + unrolled De Boor with v_rcp_f32
//   - 128-bit global loads/stores, 32 elements/thread, global_prefetch_b8
// ---------------------------------------------------------------------------

#define TKN   1024   // number of knots (power of two)
#define NC    1020   // number of coefficients = TKN - K - 1
#define KDEG  3
#define BLOCK 256
#define V4PT  8      // float4 iterations per thread -> 32 elems/thread

#if defined(__gfx1250__) && __has_builtin(__builtin_amdgcn_global_load_async_to_lds_b128)
#define HAVE_ASYNC_LDS 1
#endif

// GCC-style vector to match the builtin's parameter type exactly:
//   "__attribute__((__vector_size__(4 * sizeof(int)))) int"
typedef int v4i_b128 __attribute__((vector_size(16)));
typedef v4i_b128 __attribute__((address_space(1)))* g_b128_ptr;   // global (AS1)
typedef v4i_b128 __attribute__((address_space(3)))* l_b128_ptr;   // LDS (AS3)

__device__ __forceinline__ float fast_rcp(float v) {
#if __has_builtin(__builtin_amdgcn_rcpf)
    return __builtin_amdgcn_rcpf(v);
#else
    return 1.0f / v;
#endif
}

__device__ __forceinline__ void async_g2l_b128(const void* g, void* l) {
#if defined(HAVE_ASYNC_LDS)
    // low 32 bits of a generic LDS pointer == LDS byte address (flat aperture rule)
    __builtin_amdgcn_global_load_async_to_lds_b128(
        (g_b128_ptr)(uintptr_t)g,
        (l_b128_ptr)(uint32_t)(uintptr_t)l,
        0, 0);
#else
    (void)g; (void)l;
#endif
}

__device__ __forceinline__ void wait_async_zero() {
#if defined(HAVE_ASYNC_LDS)
#if __has_builtin(__builtin_amdgcn_s_wait_asynccnt)
    __builtin_amdgcn_s_wait_asynccnt(0);
#else
    asm volatile("s_wait_asynccnt 0" ::: "memory");
#endif
#endif
}

// ---------------------------------------------------------------------------
// Kernel 1: bitonic sort of the 1024 knots, one block of 1024 threads.
// ---------------------------------------------------------------------------
__global__ __launch_bounds__(TKN) void knot_sort_kernel(
    const float* __restrict__ t, float* __restrict__ ts_out)
{
    __shared__ float s[TKN];
    const int tid = threadIdx.x;
    s[tid] = t[tid];
    for (int k = 2; k <= TKN; k <<= 1) {
        for (int j = k >> 1; j > 0; j >>= 1) {
            __syncthreads();
            const int ixj = tid ^ j;
            if (ixj > tid) {
                const float a = s[tid];
                const float b = s[ixj];
                const bool asc = ((tid & k) == 0);
                if (asc ? (a > b) : (a < b)) { s[tid] = b; s[ixj] = a; }
            }
        }
    }
    __syncthreads();
    ts_out[tid] = s[tid];
}

// ---------------------------------------------------------------------------
// De Boor evaluation of one point against LDS-resident tables.
// ---------------------------------------------------------------------------
__device__ __forceinline__ float deboor_eval(float xv,
                                             const float* __restrict__ s_t,
                                             const float* __restrict__ s_c)
{
    // count of knots <= xv  (== searchsorted(ts, x, side='right'))
    int pos = 0;
#pragma unroll
    for (int step = TKN; step > 0; step >>= 1) {
        const int np = pos + step;
        if (np <= TKN && s_t[np - 1] <= xv) pos = np;
    }
    int i = pos - 1;
    i = (i < KDEG) ? KDEG : ((i > NC - 1) ? NC - 1 : i);

    const float t_m2 = s_t[i - 2];
    const float t_m1 = s_t[i - 1];
    const float t_0  = s_t[i];
    const float t_p1 = s_t[i + 1];
    const float t_p2 = s_t[i + 2];
    const float t_p3 = s_t[i + 3];

    float d0 = s_c[i - 3];
    float d1 = s_c[i - 2];
    float d2 = s_c[i - 1];
    float d3 = s_c[i];

    float a;
    // r = 1
    a = (xv - t_0)  * fast_rcp(t_p3 - t_0);   d3 = d2 + a * (d3 - d2);
    a = (xv - t_m1) * fast_rcp(t_p2 - t_m1);  d2 = d1 + a * (d2 - d1);
    a = (xv - t_m2) * fast_rcp(t_p1 - t_m2);  d1 = d0 + a * (d1 - d0);
    // r = 2
    a = (xv - t_0)  * fast_rcp(t_p2 - t_0);   d3 = d2 + a * (d3 - d2);
    a = (xv - t_m1) * fast_rcp(t_p1 - t_m1);  d2 = d1 + a * (d2 - d1);
    // r = 3
    a = (xv - t_0)  * fast_rcp(t_p1 - t_0);   d3 = d2 + a * (d3 - d2);
    return d3;
}

// ---------------------------------------------------------------------------
// Kernel 2: stage tables into LDS, evaluate 32 x-values per thread.
// ---------------------------------------------------------------------------
__global__ __launch_bounds__(BLOCK) void bspline_eval_kernel(
    const float* __restrict__ x,
    const float* __restrict__ ts,   // sorted knots (d_ws)
    const float* __restrict__ c,
    float* __restrict__ out,
    int n, int n4)
{
    __shared__ __align__(16) float s_t[TKN];
    __shared__ __align__(16) float s_c[NC];

    const int tid = threadIdx.x;

#if defined(HAVE_ASYNC_LDS)
    // 1024 knots = 256 x b128 (one per lane); 1020 coeffs = 255 x b128.
    {
        const float4* tsv = (const float4*)ts;
        const float4* cv  = (const float4*)c;
        async_g2l_b128(tsv + tid, ((float4*)s_t) + tid);
        if (tid < NC / 4) async_g2l_b128(cv + tid, ((float4*)s_c) + tid);
        wait_async_zero();
    }
#else
    for (int k = tid; k < TKN; k += BLOCK) s_t[k] = ts[k];
    for (int k = tid; k < NC;  k += BLOCK) s_c[k] = c[k];
#endif
    __syncthreads();

    const float4* __restrict__ x4 = (const float4*)x;
    float4* __restrict__ o4 = (float4*)out;

    const int base = blockIdx.x * (BLOCK * V4PT) + tid;

#pragma unroll
    for (int it = 0; it < V4PT; ++it) {
        const int idx = base + it * BLOCK;
        if (idx < n4) {
            if (idx + BLOCK < n4) __builtin_prefetch(&x4[idx + BLOCK], 0, 3);
            const float4 v = x4[idx];
            float4 r;
            r.x = deboor_eval(v.x, s_t, s_c);
            r.y = deboor_eval(v.y, s_t, s_c);
            r.z = deboor_eval(v.z, s_t, s_c);
            r.w = deboor_eval(v.w, s_t, s_c);
            o4[idx] = r;
        }
    }

    // scalar tail (n not divisible by 4), handled by block 0
    const int tail_start = n4 << 2;
    if (blockIdx.x == 0 && tid < (n - tail_start)) {
        const int idx = tail_start + tid;
        out[idx] = deboor_eval(x[idx], s_t, s_c);
    }
}

// ---------------------------------------------------------------------------
extern "C" void kernel_launch(void* const* d_in, const int* in_sizes, int n_in,
                              void* d_out, int out_size, void* d_ws, size_t ws_size,
                              hipStream_t stream)
{
    const float* x = (const float*)d_in[0];
    const float* t = (const float*)d_in[1];
    const float* c = (const float*)d_in[2];
    float* out = (float*)d_out;
    float* ts  = (float*)d_ws;   // sorted knots: 4 KB of scratch
    const int n = in_sizes[0];

    knot_sort_kernel<<<1, TKN, 0, stream>>>(t, ts);

    const int n4 = n >> 2;
    const int per_block = BLOCK * V4PT;            // float4s per block
    int blocks = (n4 + per_block - 1) / per_block; // 2048 for n = 16,777,216
    if (blocks < 1) blocks = 1;
    bspline_eval_kernel<<<blocks, BLOCK, 0, stream>>>(x, ts, c, out, n, n4);
}